// FusionMethodB_46703474376899
// MI455X (gfx1250) — compile-verified
//
#include <hip/hip_runtime.h>

typedef __attribute__((ext_vector_type(2))) float v2f;
typedef __attribute__((ext_vector_type(8))) float v8f;

#define DIM 128   // input feature dim per layer
#define HID 128   // output feature dim per layer

// ---------------------------------------------------------------------------
// Small utility kernels
// ---------------------------------------------------------------------------
__global__ void gcn_fill(float* p, float v, int n) {
    int i = blockIdx.x * 256 + threadIdx.x;
    if (i < n) p[i] = v;
}

__global__ void gcn_deg_scatter(float* deg, const int* __restrict__ col,
                                const float* __restrict__ w, int E) {
    int e = blockIdx.x * 256 + threadIdx.x;
    if (e < E) atomicAdd(&deg[col[e]], w[e]);
}

__global__ void gcn_to_rsqrt(float* p, int n) {
    int i = blockIdx.x * 256 + threadIdx.x;
    if (i < n) {
        float d = p[i];
        p[i] = (d > 0.0f) ? rsqrtf(d) : 0.0f;
    }
}

// ---------------------------------------------------------------------------
// Dual-weight WMMA GEMM: hWsc = h @ Wsc ; hWfc = h @ Wfc
// block = 256 threads (8 waves). Each wave computes a 16x128 output strip.
// A strip (16 rows x 128 K) is preloaded into 32 v2f fragments and reused
// across both weight phases (Wsc then Wfc staged through 64KB of LDS).
//
// LDS weight layout is K-pair interleaved:
//     lds[(k>>1)*256 + j*2 + (k&1)] = W[k][j]
// so a B fragment {W[kb][j], W[kb+1][j]} is one aligned ds_load_b64 into an
// even VGPR pair — no repack movs between LDS and v_wmma sources, and the
// read pattern is bank-conflict-free within each half-wave.
//
// WMMA layouts per CDNA5 ISA 7.12.2:
//   A 16x4 f32 : lanes 0-15 hold K=0,1 in VGPR0,1 ; lanes 16-31 hold K=2,3
//   B 4x16 f32 : VGPR0 lanes 0-15 K=0, lanes 16-31 K=2 ; VGPR1 K=1 / K=3
//   C 16x16    : VGPR r -> M=r (lanes 0-15), M=r+8 (lanes 16-31), N=lane&15
// ---------------------------------------------------------------------------
__global__ __launch_bounds__(256)
void gcn_gemm_dual_wmma(const float* __restrict__ h,
                        const float* __restrict__ Wsc,
                        const float* __restrict__ Wfc,
                        float* __restrict__ hWsc,
                        float* __restrict__ hWfc,
                        int N) {
    __shared__ float lds[DIM * HID];   // 64 KB weight stage (interleaved)

    const int tid  = threadIdx.x;
    const int wave = tid >> 5;
    const int lane = tid & 31;
    const int c    = lane & 15;        // N column within 16-wide tile
    const int g    = lane >> 4;        // half-wave selector (K pairs)

    const int rowBase = blockIdx.x * 128 + wave * 16;
    int r  = rowBase + c;
    int rl = (r < N) ? r : (N - 1);    // clamp, keep EXEC all-ones for WMMA

    // Preload A strip: a[kk] holds h[rl][4*kk + 2g + {0,1}]
    v2f a[32];
    const float* hrow = h + (size_t)rl * DIM + 2 * g;
#pragma unroll
    for (int kk = 0; kk < 32; ++kk)
        a[kk] = *(const v2f*)(hrow + kk * 4);

#pragma unroll 1
    for (int phase = 0; phase < 2; ++phase) {
        const float* W   = phase ? Wfc : Wsc;
        float*       out = phase ? hWfc : hWsc;

        __syncthreads();   // protect LDS from previous phase readers
        // cooperative weight load: global float4 reads, interleaved LDS stores
        for (int t = tid; t < (DIM * HID) / 4; t += 256) {
            float4 w4 = ((const float4*)W)[t];
            int k  = t >> 5;            // row of W (K index), 32 float4 per row
            int j0 = (t & 31) * 4;      // starting output column
            float* dst = lds + (k >> 1) * 256 + (k & 1);
            dst[(j0 + 0) * 2] = w4.x;
            dst[(j0 + 1) * 2] = w4.y;
            dst[(j0 + 2) * 2] = w4.z;
            dst[(j0 + 3) * 2] = w4.w;
        }
        __syncthreads();

        v8f acc[8] = {};
#pragma unroll
        for (int kk = 0; kk < 32; ++kk) {
            const int kb = kk * 4 + 2 * g;             // even K base for this half
            const float* bsrc = lds + (kb >> 1) * 256; // + j*2 per lane below
#pragma unroll
            for (int n = 0; n < 8; ++n) {
                v2f b = *(const v2f*)(bsrc + (n * 16 + c) * 2);
                acc[n] = __builtin_amdgcn_wmma_f32_16x16x4_f32(
                    false, a[kk], false, b, (short)0, acc[n], false, false);
            }
        }

        // store C tiles: lane (c,g), VGPR vr -> row rowBase + vr + 8*g
#pragma unroll
        for (int vr = 0; vr < 8; ++vr) {
            int m = rowBase + vr + 8 * g;
            if (m < N) {
                float* op = out + (size_t)m * HID + c;
#pragma unroll
                for (int n = 0; n < 8; ++n)
                    op[n * 16] = acc[n][vr];
            }
        }
    }
}

// ---------------------------------------------------------------------------
// out = b_sc + b_fc + dis_sc^2 * hWsc + dis_fc^2 * hWfc   (self-loop terms)
// ---------------------------------------------------------------------------
__global__ void gcn_init_out(float* __restrict__ out,
                             const float* __restrict__ bsc,
                             const float* __restrict__ bfc,
                             const float* __restrict__ dsc,
                             const float* __restrict__ dfc,
                             const float* __restrict__ hWsc,
                             const float* __restrict__ hWfc,
                             int total) {
    int i = blockIdx.x * 256 + threadIdx.x;
    if (i >= total) return;
    int v = i >> 7;
    int j = i & 127;
    float a = dsc[v];
    float b = dfc[v];
    out[i] = bsc[j] + bfc[j] + a * a * hWsc[i] + b * b * hWfc[i];
}

// ---------------------------------------------------------------------------
// Edge scatter: 128 threads per edge, one feature each.
// out[col] += dis[row]*w*dis[col] * hW[row]
// ---------------------------------------------------------------------------
__global__ void gcn_edge_scatter(float* __restrict__ out,
                                 const float* __restrict__ hW,
                                 const int* __restrict__ row,
                                 const int* __restrict__ col,
                                 const float* __restrict__ w,
                                 const float* __restrict__ dis,
                                 int E) {
    int idx = blockIdx.x * 256 + threadIdx.x;
    int e = idx >> 7;
    if (e >= E) return;
    int j = idx & 127;
    int r = row[e];
    int d = col[e];
    float norm = dis[r] * w[e] * dis[d];
    atomicAdd(&out[(size_t)d * HID + j], hW[(size_t)r * HID + j] * norm);
}

// ---------------------------------------------------------------------------
// BatchNorm statistics: per-feature sum and sum-of-squares.
// ---------------------------------------------------------------------------
__global__ __launch_bounds__(256)
void gcn_bn_stats(const float* __restrict__ x, float* __restrict__ stats, int N) {
    __shared__ float ls[256];
    __shared__ float lq[256];
    int tid  = threadIdx.x;
    int j    = tid & 127;
    int half = tid >> 7;
    float s = 0.0f, q = 0.0f;
    for (int v = blockIdx.x * 2 + half; v < N; v += gridDim.x * 2) {
        float val = x[(size_t)v * HID + j];
        s += val;
        q += val * val;
    }
    ls[tid] = s;
    lq[tid] = q;
    __syncthreads();
    if (half == 0) {
        atomicAdd(&stats[j],       s + ls[tid + 128]);
        atomicAdd(&stats[128 + j], q + lq[tid + 128]);
    }
}

__global__ void gcn_bn_finalize(const float* __restrict__ stats,
                                const float* __restrict__ gamma,
                                const float* __restrict__ beta,
                                float* __restrict__ coef, int N) {
    int j = threadIdx.x;   // 128 threads
    float invN = 1.0f / (float)N;
    float mean = stats[j] * invN;
    float var  = stats[128 + j] * invN - mean * mean;
    float inv  = rsqrtf(var + 1e-5f);
    float sc   = gamma[j] * inv;
    coef[j]       = sc;
    coef[128 + j] = beta[j] - mean * sc;
}

__global__ void gcn_bn_apply_relu(float* __restrict__ x,
                                  const float* __restrict__ coef, int total) {
    int i = blockIdx.x * 256 + threadIdx.x;
    if (i >= total) return;
    int j = i & 127;
    float y = x[i] * coef[j] + coef[128 + j];
    x[i] = fmaxf(y, 0.0f);
}

// ---------------------------------------------------------------------------
// Host-side launch
// ---------------------------------------------------------------------------
extern "C" void kernel_launch(void* const* d_in, const int* in_sizes, int n_in,
                              void* d_out, int out_size, void* d_ws, size_t ws_size,
                              hipStream_t stream) {
    const float* x     = (const float*)d_in[0];
    const int*   eisc  = (const int*)d_in[1];
    const float* ewsc  = (const float*)d_in[2];
    const int*   eifc  = (const int*)d_in[3];
    const float* ewfc  = (const float*)d_in[4];
    const float* Wsc   = (const float*)d_in[5];
    const float* bsc   = (const float*)d_in[6];
    const float* Wfc   = (const float*)d_in[7];
    const float* bfc   = (const float*)d_in[8];
    const float* gamma = (const float*)d_in[9];
    const float* beta  = (const float*)d_in[10];

    const int N = in_sizes[0] / DIM;
    const int E = in_sizes[2];
    const int L = in_sizes[6] / HID;

    // workspace layout (floats)
    float* ws    = (float*)d_ws;
    float* hWsc  = ws;
    float* hWfc  = hWsc + (size_t)N * HID;
    float* Obuf  = hWfc + (size_t)N * HID;
    float* dsc   = Obuf + (size_t)N * HID;
    float* dfc   = dsc + N;
    float* stats = dfc + N;      // 256 floats
    float* coef  = stats + 256;  // 256 floats

    const int NB256  = (N + 255) / 256;
    const int EB256  = (E + 255) / 256;
    const int NH     = N * HID;
    const int NHB    = (NH + 255) / 256;
    const int EFB    = (int)(((size_t)E * HID + 255) / 256);
    const int GEMMB  = (N + 127) / 128;

    // ---- degree / normalization factors (layer-invariant) ----
    gcn_fill<<<NB256, 256, 0, stream>>>(dsc, 1.0f, N);   // self-loop weight 1
    gcn_fill<<<NB256, 256, 0, stream>>>(dfc, 1.0f, N);
    gcn_deg_scatter<<<EB256, 256, 0, stream>>>(dsc, eisc + E, ewsc, E);
    gcn_deg_scatter<<<EB256, 256, 0, stream>>>(dfc, eifc + E, ewfc, E);
    gcn_to_rsqrt<<<NB256, 256, 0, stream>>>(dsc, N);
    gcn_to_rsqrt<<<NB256, 256, 0, stream>>>(dfc, N);

    const float* hin = x;
    for (int i = 0; i < L; ++i) {
        float* outb = (i == L - 1) ? (float*)d_out : Obuf;
        const size_t wo = (size_t)i * DIM * HID;

        gcn_gemm_dual_wmma<<<GEMMB, 256, 0, stream>>>(
            hin, Wsc + wo, Wfc + wo, hWsc, hWfc, N);

        gcn_init_out<<<NHB, 256, 0, stream>>>(
            outb, bsc + i * HID, bfc + i * HID, dsc, dfc, hWsc, hWfc, NH);

        gcn_edge_scatter<<<EFB, 256, 0, stream>>>(
            outb, hWsc, eisc, eisc + E, ewsc, dsc, E);
        gcn_edge_scatter<<<EFB, 256, 0, stream>>>(
            outb, hWfc, eifc, eifc + E, ewfc, dfc, E);

        gcn_fill<<<1, 256, 0, stream>>>(stats, 0.0f, 256);
        gcn_bn_stats<<<256, 256, 0, stream>>>(outb, stats, N);
        gcn_bn_finalize<<<1, 128, 0, stream>>>(
            stats, gamma + i * HID, beta + i * HID, coef, N);
        gcn_bn_apply_relu<<<NHB, 256, 0, stream>>>(outb, coef, NH);

        hin = outb;
    }
}